// Encoder2_45913200394640
// MI455X (gfx1250) — compile-verified
//
#include <hip/hip_runtime.h>
#include <math.h>

#define N_FEAT 128

typedef __attribute__((ext_vector_type(2))) float v2f;
typedef __attribute__((ext_vector_type(8))) float v8f;

// ---------------- degree / dinv ----------------
__global__ void deg_init_kernel(float* __restrict__ deg, int n) {
    int i = blockIdx.x * blockDim.x + threadIdx.x;
    if (i < n) deg[i] = 1.0f;               // self-loop: deg starts at 1
}

__global__ void deg_scatter_kernel(const int* __restrict__ dst, float* __restrict__ deg, int E) {
    int e = blockIdx.x * blockDim.x + threadIdx.x;
    if (e < E) atomicAdd(&deg[dst[e]], 1.0f);
}

__global__ void deg_rsqrt_kernel(float* __restrict__ deg, int n) {
    int i = blockIdx.x * blockDim.x + threadIdx.x;
    if (i < n) deg[i] = rsqrtf(deg[i]);     // deg >= 1 always
}

// ---------------- dense GEMM: H[M x 128] = X[M x 128] @ W[128 x 128] ----------------
// One wave per 16x16 output tile, full f32 precision via V_WMMA_F32_16X16X4_F32.
// Block = 256 threads = 8 waves; wave w covers N-tile w (128 cols = 8 tiles);
// blockIdx.x = M-tile. All 8 waves share the same 16 rows of X (cache-friendly).
__global__ void gemm_wmma_kernel(const float* __restrict__ X, const float* __restrict__ W,
                                 float* __restrict__ H, int M) {
    const int wave = threadIdx.x >> 5;
    const int lane = threadIdx.x & 31;
    const int half = lane >> 4;   // 0: K pair {0,1}; 1: K pair {2,3}
    const int l16  = lane & 15;
    const int row0 = blockIdx.x * 16;
    if (row0 >= M) return;
    const int col = wave * 16 + l16;

    v8f acc = {};
    const float* xrow = X + (size_t)(row0 + l16) * N_FEAT + half * 2;
    #pragma unroll
    for (int k0 = 0; k0 < N_FEAT; k0 += 4) {
        v2f a;  // A 16x4: lane l16 = row M, VGPR0/1 = K = k0+half*2 (+1)
        a.x = xrow[k0 + 0];
        a.y = xrow[k0 + 1];
        v2f b;  // B 4x16: lane l16 = col N, VGPR0/1 = K = k0+half*2 (+1)
        b.x = W[(size_t)(k0 + half * 2 + 0) * N_FEAT + col];
        b.y = W[(size_t)(k0 + half * 2 + 1) * N_FEAT + col];
        acc = __builtin_amdgcn_wmma_f32_16x16x4_f32(
            /*neg_a=*/false, a, /*neg_b=*/false, b,
            /*c_mod=*/(short)0, acc, /*reuse_a=*/false, /*reuse_b=*/false);
    }
    // D 16x16 f32: VGPR r -> M = r + half*8, N = l16
    float* out = H + (size_t)row0 * N_FEAT + wave * 16 + l16;
    #pragma unroll
    for (int r = 0; r < 8; ++r) {
        out[(size_t)(half * 8 + r) * N_FEAT] = acc[r];
    }
}

// ---------------- zero buffer ----------------
__global__ void zero_kernel(float4* __restrict__ p, long long n4) {
    long long i = (long long)blockIdx.x * blockDim.x + threadIdx.x;
    if (i < n4) p[i] = make_float4(0.f, 0.f, 0.f, 0.f);
}

// ---------------- edge scatter: agg[dst] += h[src] * dinv[src]*dinv[dst] ----------------
// One wave (32 lanes) per edge; lane handles a float4 slice of the 128-wide feature row.
__global__ void scatter_kernel(const float* __restrict__ H, const float* __restrict__ dinv,
                               const int* __restrict__ src, const int* __restrict__ dst,
                               float* __restrict__ agg, int E) {
    const int lane = threadIdx.x & 31;
    const int e = blockIdx.x * (blockDim.x >> 5) + (threadIdx.x >> 5);
    if (e >= E) return;
    const int s = src[e];          // wave-uniform (scalar load)
    const int d = dst[e];
    const float nrm = dinv[s] * dinv[d];
    const float4 v = *(const float4*)(H + (size_t)s * N_FEAT + lane * 4);
    float* p = agg + (size_t)d * N_FEAT + lane * 4;
    atomicAdd(p + 0, v.x * nrm);
    atomicAdd(p + 1, v.y * nrm);
    atomicAdd(p + 2, v.z * nrm);
    atomicAdd(p + 3, v.w * nrm);
}

// ---------------- finalize: out = prelu(agg + h*dinv^2 + bias, alpha) ----------------
// One wave per node; may run in place (out == agg).
__global__ void finalize_kernel(const float* __restrict__ agg, const float* __restrict__ H,
                                const float* __restrict__ dinv, const float* __restrict__ bias,
                                const float* __restrict__ alpha, float* __restrict__ out, int n) {
    const int lane = threadIdx.x & 31;
    const int node = blockIdx.x * (blockDim.x >> 5) + (threadIdx.x >> 5);
    if (node >= n) return;
    const float di = dinv[node];
    const float di2 = di * di;
    const int f0 = lane * 4;
    const float4 ag = *(const float4*)(agg + (size_t)node * N_FEAT + f0);
    const float4 hv = *(const float4*)(H + (size_t)node * N_FEAT + f0);
    const float4 bv = *(const float4*)(bias + f0);
    const float4 av = *(const float4*)(alpha + f0);
    float4 r;
    r.x = ag.x + hv.x * di2 + bv.x;  r.x = r.x > 0.f ? r.x : av.x * r.x;
    r.y = ag.y + hv.y * di2 + bv.y;  r.y = r.y > 0.f ? r.y : av.y * r.y;
    r.z = ag.z + hv.z * di2 + bv.z;  r.z = r.z > 0.f ? r.z : av.z * r.z;
    r.w = ag.w + hv.w * di2 + bv.w;  r.w = r.w > 0.f ? r.w : av.w * r.w;
    *(float4*)(out + (size_t)node * N_FEAT + f0) = r;
}

extern "C" void kernel_launch(void* const* d_in, const int* in_sizes, int n_in,
                              void* d_out, int out_size, void* d_ws, size_t ws_size,
                              hipStream_t stream) {
    (void)n_in; (void)out_size; (void)ws_size;
    const float* x  = (const float*)d_in[0];
    const int*   ei = (const int*)d_in[1];   // [2, E] edge index (harness: integer -> int32)
    const float* W1 = (const float*)d_in[2];
    const float* b1 = (const float*)d_in[3];
    const float* W2 = (const float*)d_in[4];
    const float* b2 = (const float*)d_in[5];
    const float* al = (const float*)d_in[6];

    const int N = in_sizes[0] / N_FEAT;      // 100000
    const int E = in_sizes[1] / 2;           // 1600000
    const int* src = ei;
    const int* dst = ei + E;

    // workspace layout: [dinv: N floats][hbuf: N*128 floats]
    float* dinv = (float*)d_ws;
    size_t off = (((size_t)N * sizeof(float)) + 255) & ~(size_t)255;
    float* hbuf = (float*)((char*)d_ws + off);
    float* agg  = (float*)d_out;             // d_out doubles as agg / z1 scratch
    float* outp = (float*)d_out;

    // ---- degrees -> dinv (once; deg is identical for both layers) ----
    deg_init_kernel<<<(N + 255) / 256, 256, 0, stream>>>(dinv, N);
    deg_scatter_kernel<<<(E + 255) / 256, 256, 0, stream>>>(dst, dinv, E);
    deg_rsqrt_kernel<<<(N + 255) / 256, 256, 0, stream>>>(dinv, N);

    const int mTiles = (N + 15) / 16;               // 6250 blocks (8 waves each)
    const long long n4 = (long long)N * N_FEAT / 4; // float4 count
    const int zBlocks = (int)((n4 + 255) / 256);
    const int eBlocks = (E + 7) / 8;                // 8 edge-waves per block
    const int nBlocks = (N + 7) / 8;                // 8 node-waves per block

    // ---- layer 1 ----
    gemm_wmma_kernel<<<mTiles, 256, 0, stream>>>(x, W1, hbuf, N);
    zero_kernel<<<zBlocks, 256, 0, stream>>>((float4*)agg, n4);
    scatter_kernel<<<eBlocks, 256, 0, stream>>>(hbuf, dinv, src, dst, agg, E);
    finalize_kernel<<<nBlocks, 256, 0, stream>>>(agg, hbuf, dinv, b1, al, outp, N);

    // ---- layer 2 (z1 lives in d_out; gemm2 consumes it before d_out is re-zeroed) ----
    gemm_wmma_kernel<<<mTiles, 256, 0, stream>>>(outp, W2, hbuf, N);
    zero_kernel<<<zBlocks, 256, 0, stream>>>((float4*)agg, n4);
    scatter_kernel<<<eBlocks, 256, 0, stream>>>(hbuf, dinv, src, dst, agg, E);
    finalize_kernel<<<nBlocks, 256, 0, stream>>>(agg, hbuf, dinv, b2, al, outp, N);
}